// HEALPix_UNet_Processor_85023172592645
// MI455X (gfx1250) — compile-verified
//
#include <hip/hip_runtime.h>
#include <hip/hip_bf16.h>

// ---------------------------------------------------------------------------
// HEALPix graph U-Net, MI455X (gfx1250, wave32).
// All dense GEMMs: V_WMMA_F32_16X16X4_F32 (exact fp32, D = A*B + C), K=128
// panels, weights pre-packed to WMMA fragment order (b128 loads), A tiles
// staged into LDS via GLOBAL_LOAD_ASYNC_TO_LDS_B128 (ASYNCcnt path).
// ---------------------------------------------------------------------------

typedef float v2f __attribute__((ext_vector_type(2)));
typedef float v4f __attribute__((ext_vector_type(4)));
typedef float v8f __attribute__((ext_vector_type(8)));

#define Dm    128
#define MBAT  24          // B*C*T = 2*1*12
#define NN4   3072
#define NN3   768
#define NN2   192
#define APITCH 132        // LDS row pitch (dwords): 128 + 4 pad -> no bank conflicts
#define ASEC  (16 * APITCH)

// --------------------------------------------------------------------------
// SpMM: out[m,n,:] = alpha * sum_{j<8} v[8n+j] * in[m, c[8n+j], :]
//                    + beta * aux[m,n,:]          (aux may be null)
// np.nonzero edge lists: exactly 8 sorted cols per row, row n owns [8n,8n+8).
// --------------------------------------------------------------------------
__global__ __launch_bounds__(128) void spmm_kernel(
    const float* __restrict__ in, const int* __restrict__ cidx,
    const float* __restrict__ val, const float* __restrict__ aux,
    float alpha, float beta, float* __restrict__ out, int n) {
  const int r = blockIdx.x;
  const int d = threadIdx.x;
  const int m = r / n;
  const int nn = r - m * n;
  const float* __restrict__ base = in + (size_t)m * n * Dm;
  float s = 0.f;
#pragma unroll
  for (int j = 0; j < 8; ++j) {
    const int e = nn * 8 + j;
    s += val[e] * base[(size_t)cidx[e] * Dm + d];
  }
  float o = alpha * s;
  if (aux) o += beta * aux[(size_t)r * Dm + d];
  out[(size_t)r * Dm + d] = o;
}

// --------------------------------------------------------------------------
// Mean-pool 4 nested children -> parent (p = n/4, matching arange(n)//4).
// --------------------------------------------------------------------------
__global__ __launch_bounds__(128) void pool_kernel(
    const float* __restrict__ in, float* __restrict__ out, int n_lo) {
  const int r = blockIdx.x;
  const int d = threadIdx.x;
  const int m = r / n_lo;
  const int p = r - m * n_lo;
  const float* __restrict__ base = in + ((size_t)m * n_lo + p) * 4 * Dm;
  out[(size_t)r * Dm + d] =
      0.25f * (base[d] + base[Dm + d] + base[2 * Dm + d] + base[3 * Dm + d]);
}

// --------------------------------------------------------------------------
// Weight pack: dst[((c*128+col)*2+h)*4+q] = src[K*kStride + col*cStride + kOff]
// with K = 8c + 2h + 4*(q>>1) + (q&1)   (WMMA f32 16x16x4 B-fragment order,
// K-chunks of 8 -> one b128 load feeds two WMMA steps).
//   cheb section k: kStride=3*Dm, cStride=1,  kOff=k*Dm   (B = W[(d*3+k)][col])
//   pointwise:      kStride=1,    cStride=Dm, kOff=0      (B = W^T)
// --------------------------------------------------------------------------
__global__ __launch_bounds__(256) void pack_w_kernel(
    const float* __restrict__ src, float* __restrict__ dst,
    int kStride, int cStride, int kOff) {
  const int idx = blockIdx.x * 256 + threadIdx.x;   // 0 .. 16*128*8-1
  const int q = idx & 3;
  const int h = (idx >> 2) & 1;
  const int col = (idx >> 3) & 127;
  const int c = idx >> 10;
  const int K = 8 * c + 2 * h + 4 * (q >> 1) + (q & 1);
  dst[idx] = src[(size_t)K * kStride + (size_t)col * cStride + kOff];
}

// --------------------------------------------------------------------------
// Async stage of one 16x128 A tile into LDS (pitch APITCH) via the CDNA5
// async-to-LDS path. 512 x 16B chunks; each of 256 threads issues 2.
// --------------------------------------------------------------------------
__device__ __forceinline__ void stage_tile_gather(
    float* As, int sec, const float* __restrict__ src,
    const int* rows /*LDS, 16 entries*/, int tid) {
#pragma unroll
  for (int t = 0; t < 2; ++t) {
    const int ch = tid + t * 256;              // 0..511
    const int i = ch >> 5;                     // tile row 0..15
    const int j = ch & 31;                     // 16B chunk within row
    const unsigned lds =
        (unsigned)(uintptr_t)(const void*)&As[sec * ASEC + i * APITCH + j * 4];
    const void* gp = (const void*)(src + (size_t)rows[i] * Dm + j * 4);
    asm volatile("global_load_async_to_lds_b128 %0, %1, off"
                 :: "v"(lds), "v"(gp) : "memory");
  }
}

__device__ __forceinline__ void stage_tile_ident(
    float* As, int sec, const float* __restrict__ src, int rowbase, int tid) {
#pragma unroll
  for (int t = 0; t < 2; ++t) {
    const int ch = tid + t * 256;
    const int i = ch >> 5;
    const int j = ch & 31;
    const unsigned lds =
        (unsigned)(uintptr_t)(const void*)&As[sec * ASEC + i * APITCH + j * 4];
    const void* gp = (const void*)(src + (size_t)(rowbase + i) * Dm + j * 4);
    asm volatile("global_load_async_to_lds_b128 %0, %1, off"
                 :: "v"(lds), "v"(gp) : "memory");
  }
}

// --------------------------------------------------------------------------
// K=128 accumulation panel: 16 chunks x (2 WMMA per chunk).
// Per chunk: 1 global_load_b128 (packed B) + 2 ds_load_b64 (A) + 2 WMMA.
// --------------------------------------------------------------------------
__device__ __forceinline__ v8f accum128(v8f acc, const float* As, int sec,
                                        const float* __restrict__ Bp,
                                        int col, int hl, int arow) {
  const float* Arow = As + sec * ASEC + arow * APITCH;
#pragma unroll 4
  for (int c = 0; c < 16; ++c) {
    const v4f bv = *(const v4f*)(Bp + (((size_t)c * Dm + col) * 2 + hl) * 4);
    const v2f a0 = *(const v2f*)(Arow + c * 8 + hl * 2);
    const v2f a1 = *(const v2f*)(Arow + c * 8 + 4 + hl * 2);
    const v2f b0 = {bv.x, bv.y};
    const v2f b1 = {bv.z, bv.w};
    acc = __builtin_amdgcn_wmma_f32_16x16x4_f32(false, a0, false, b0,
                                                (short)0, acc, false, false);
    acc = __builtin_amdgcn_wmma_f32_16x16x4_f32(false, a1, false, b1,
                                                (short)0, acc, false, false);
  }
  return acc;
}

// --------------------------------------------------------------------------
// Unified GEMM: out[r,:] = sum_s a_s[g_s(r),:] @ Bp_s + bias (+ skip[r,:])
//   source 0 optionally row-gathered through pmap (up-sample); sources 1/2
//   (Chebyshev x1/x2) identity. Block = 16 rows; wave w owns cols [16w,16w+16).
// In-place safe for out == a0 (tile rows fully staged before writes).
// --------------------------------------------------------------------------
__global__ __launch_bounds__(256) void gemm128_kernel(
    const float* __restrict__ a0, const float* __restrict__ Bp0,
    const float* __restrict__ a1, const float* __restrict__ Bp1,
    const float* __restrict__ a2, const float* __restrict__ Bp2,
    const int* __restrict__ pmap, const float* __restrict__ bias,
    const float* __restrict__ skip, float* __restrict__ out,
    int n_out, int n_in) {
  __shared__ float As[3 * ASEC];     // 3 x 16 x 132 dwords = 24.75 KB
  __shared__ int srcr[16];
  const int tid = threadIdx.x;
  const int rowbase = blockIdx.x * 16;

  for (int f = tid; f < 2 * Dm; f += 256)   // packed-B prefetch (L2 resident)
    __builtin_prefetch(&Bp0[(size_t)f * 64], 0, 3);

  if (tid < 16) {
    const int r = rowbase + tid;
    const int m = r / n_out;
    const int n = r - m * n_out;
    const int sn = pmap ? pmap[n] : n;
    srcr[tid] = m * n_in + sn;
  }
  __syncthreads();

  stage_tile_gather(As, 0, a0, srcr, tid);
  if (a1) stage_tile_ident(As, 1, a1, rowbase, tid);
  if (a2) stage_tile_ident(As, 2, a2, rowbase, tid);
  asm volatile("s_wait_asynccnt 0x0" ::: "memory");
  __syncthreads();

  const int lane = tid & 31;
  const int wave = tid >> 5;
  const int arow = lane & 15;
  const int hl = lane >> 4;
  const int col = wave * 16 + arow;

  v8f acc = {};
  acc = accum128(acc, As, 0, Bp0, col, hl, arow);
  if (a1) acc = accum128(acc, As, 1, Bp1, col, hl, arow);
  if (a2) acc = accum128(acc, As, 2, Bp2, col, hl, arow);

  const float bv = bias[col];
#pragma unroll
  for (int i = 0; i < 8; ++i) {
    const int row = rowbase + i + hl * 8;
    float o = acc[i] + bv;
    if (skip) o += skip[(size_t)row * Dm + col];
    out[(size_t)row * Dm + col] = o;
  }
}

// --------------------------------------------------------------------------
extern "C" void kernel_launch(void* const* d_in, const int* in_sizes, int n_in,
                              void* d_out, int out_size, void* d_ws, size_t ws_size,
                              hipStream_t stream) {
  const float* x   = (const float*)d_in[0];
  const int*   c4  = (const int*)  d_in[2];
  const float* v4  = (const float*)d_in[3];
  const int*   c3  = (const int*)  d_in[5];
  const float* v3  = (const float*)d_in[6];
  const int*   c2  = (const int*)  d_in[8];
  const float* v2  = (const float*)d_in[9];
  const int*   p4  = (const int*)  d_in[10];
  const int*   p3  = (const int*)  d_in[11];
  const float* W4e = (const float*)d_in[12];
  const float* W3e = (const float*)d_in[13];
  const float* Wb  = (const float*)d_in[14];
  const float* W3d = (const float*)d_in[15];
  const float* W4d = (const float*)d_in[16];
  const float* Wd43= (const float*)d_in[17];
  const float* Wd32= (const float*)d_in[18];
  const float* Wu23= (const float*)d_in[19];
  const float* Wu34= (const float*)d_in[20];
  const float* b4e = (const float*)d_in[21];
  const float* b3e = (const float*)d_in[22];
  const float* bb  = (const float*)d_in[23];
  const float* b3d = (const float*)d_in[24];
  const float* b4d = (const float*)d_in[25];
  const float* bd43= (const float*)d_in[26];
  const float* bd32= (const float*)d_in[27];
  const float* bu23= (const float*)d_in[28];
  const float* bu34= (const float*)d_in[29];
  float* out = (float*)d_out;

  const size_t S4 = (size_t)MBAT * NN4 * Dm;
  const size_t S3 = (size_t)MBAT * NN3 * Dm;
  const size_t S2 = (size_t)MBAT * NN2 * Dm;
  const size_t PW = (size_t)16 * Dm * 8;      // 16384 floats per packed panel

  float* ws  = (float*)d_ws;
  float* T4a = ws;            // x1 @ level 4
  float* T4b = T4a + S4;      // x2 @ level 4
  float* H4E = T4b + S4;      // encoder level-4 output (skip)
  float* T4c = H4E + S4;      // decoder level-4 input
  float* T3a = T4c + S4;      // x1 @ level 3
  float* T3b = T3a + S3;      // x2 @ level 3
  float* H3E = T3b + S3;      // encoder level-3 output (skip)
  float* T3c = H3E + S3;      // d3 / h3in / u3in / u3 chain
  float* T2a = T3c + S3;      // d2 / h2in
  float* T2b = T2a + S2;      // x1 @ level 2
  float* T2c = T2b + S2;      // x2 @ level 2
  float* T2d = T2c + S2;      // bottleneck output hb
  float* PK  = T2d + S2;      // packed weights
  float* P4e = PK;            // 3 panels each for the 5 cheb weights
  float* P3e = P4e + 3 * PW;
  float* Pb  = P3e + 3 * PW;
  float* P3d = Pb  + 3 * PW;
  float* P4d = P3d + 3 * PW;
  float* Pd43= P4d + 3 * PW;  // 1 panel each for the 4 pointwise weights
  float* Pd32= Pd43 + PW;
  float* Pu23= Pd32 + PW;
  float* Pu34= Pu23 + PW;

  const dim3 blkS(128), blkG(256);
  const int R4 = MBAT * NN4, R3 = MBAT * NN3, R2 = MBAT * NN2;
  const int packGrid = (int)(PW / 256);       // 64 blocks

  // ---- pre-pack weights into WMMA B-fragment order ---------------------
  const float* chebW[5] = {W4e, W3e, Wb, W3d, W4d};
  float* chebP[5] = {P4e, P3e, Pb, P3d, P4d};
  for (int wgt = 0; wgt < 5; ++wgt)
    for (int k = 0; k < 3; ++k)
      pack_w_kernel<<<packGrid, blkG, 0, stream>>>(
          chebW[wgt], chebP[wgt] + (size_t)k * PW, 3 * Dm, 1, k * Dm);
  const float* pwW[4] = {Wd43, Wd32, Wu23, Wu34};
  float* pwP[4] = {Pd43, Pd32, Pu23, Pu34};
  for (int wgt = 0; wgt < 4; ++wgt)
    pack_w_kernel<<<packGrid, blkG, 0, stream>>>(pwW[wgt], pwP[wgt], 1, Dm, 0);

  // ---- encoder level 4 --------------------------------------------------
  spmm_kernel<<<R4, blkS, 0, stream>>>(x,   c4, v4, nullptr, 1.f,  0.f, T4a, NN4);
  spmm_kernel<<<R4, blkS, 0, stream>>>(T4a, c4, v4, x,       2.f, -1.f, T4b, NN4);
  gemm128_kernel<<<R4 / 16, blkG, 0, stream>>>(
      x, P4e, T4a, P4e + PW, T4b, P4e + 2 * PW, nullptr, b4e, nullptr,
      H4E, NN4, NN4);

  // ---- down 4 -> 3, encoder level 3 ------------------------------------
  pool_kernel<<<R3, blkS, 0, stream>>>(H4E, T3c, NN3);
  gemm128_kernel<<<R3 / 16, blkG, 0, stream>>>(
      T3c, Pd43, nullptr, nullptr, nullptr, nullptr, nullptr, bd43, nullptr,
      T3c, NN3, NN3);
  spmm_kernel<<<R3, blkS, 0, stream>>>(T3c, c3, v3, nullptr, 1.f,  0.f, T3a, NN3);
  spmm_kernel<<<R3, blkS, 0, stream>>>(T3a, c3, v3, T3c,     2.f, -1.f, T3b, NN3);
  gemm128_kernel<<<R3 / 16, blkG, 0, stream>>>(
      T3c, P3e, T3a, P3e + PW, T3b, P3e + 2 * PW, nullptr, b3e, nullptr,
      H3E, NN3, NN3);

  // ---- down 3 -> 2, bottleneck -----------------------------------------
  pool_kernel<<<R2, blkS, 0, stream>>>(H3E, T2a, NN2);
  gemm128_kernel<<<R2 / 16, blkG, 0, stream>>>(
      T2a, Pd32, nullptr, nullptr, nullptr, nullptr, nullptr, bd32, nullptr,
      T2a, NN2, NN2);
  spmm_kernel<<<R2, blkS, 0, stream>>>(T2a, c2, v2, nullptr, 1.f,  0.f, T2b, NN2);
  spmm_kernel<<<R2, blkS, 0, stream>>>(T2b, c2, v2, T2a,     2.f, -1.f, T2c, NN2);
  gemm128_kernel<<<R2 / 16, blkG, 0, stream>>>(
      T2a, Pb, T2b, Pb + PW, T2c, Pb + 2 * PW, nullptr, bb, nullptr,
      T2d, NN2, NN2);

  // ---- up 2 -> 3 (+skip), decoder level 3 ------------------------------
  gemm128_kernel<<<R3 / 16, blkG, 0, stream>>>(
      T2d, Pu23, nullptr, nullptr, nullptr, nullptr, p3, bu23, H3E,
      T3c, NN3, NN2);
  spmm_kernel<<<R3, blkS, 0, stream>>>(T3c, c3, v3, nullptr, 1.f,  0.f, T3a, NN3);
  spmm_kernel<<<R3, blkS, 0, stream>>>(T3a, c3, v3, T3c,     2.f, -1.f, T3b, NN3);
  gemm128_kernel<<<R3 / 16, blkG, 0, stream>>>(
      T3c, P3d, T3a, P3d + PW, T3b, P3d + 2 * PW, nullptr, b3d, nullptr,
      T3c, NN3, NN3);

  // ---- up 3 -> 4 (+skip), decoder level 4 -> output --------------------
  gemm128_kernel<<<R4 / 16, blkG, 0, stream>>>(
      T3c, Pu34, nullptr, nullptr, nullptr, nullptr, p4, bu34, H4E,
      T4c, NN4, NN3);
  spmm_kernel<<<R4, blkS, 0, stream>>>(T4c, c4, v4, nullptr, 1.f,  0.f, T4a, NN4);
  spmm_kernel<<<R4, blkS, 0, stream>>>(T4a, c4, v4, T4c,     2.f, -1.f, T4b, NN4);
  gemm128_kernel<<<R4 / 16, blkG, 0, stream>>>(
      T4c, P4d, T4a, P4d + PW, T4b, P4d + 2 * PW, nullptr, b4d, nullptr,
      out, NN4, NN4);
}